// Convex_f_18631568130503
// MI455X (gfx1250) — compile-verified
//
#include <hip/hip_runtime.h>
#include <cstdint>
#include <cstddef>

#define B_ 256
#define N_ 8192
#define K_ 16
#define TILE_N 64
#define TILES_PER_B (N_ / TILE_N)   // 128 tiles per batch slice

typedef float f4 __attribute__((ext_vector_type(4)));

// CDNA5 async DMA: global memory -> LDS, 16 bytes per lane, tracked by ASYNCcnt.
// GVS addressing: mem = SADDR(64b) + VADDR(32b byte offset); VDST = LDS byte address.
__device__ __forceinline__ void async_load_b128(uint32_t lds_off, uint32_t voff, uint64_t sbase) {
    asm volatile("global_load_async_to_lds_b128 %0, %1, %2"
                 :: "v"(lds_off), "v"(voff), "s"(sbase)
                 : "memory");
}

__device__ __forceinline__ float convex_elem(float xm, float pm,
                                             float xc, float pc,
                                             float xp, float pp) {
    float yc  = xc + pc;
    float mid = 0.5f * ((xm + pm) + (xp + pp));
    return __builtin_fminf(yc, mid) - pc;   // == y - dy - param for interior rows
}

__global__ __launch_bounds__(256)
void convex_stencil_kernel(const float* __restrict__ x,
                           const float* __restrict__ p,
                           float* __restrict__ out) {
    // Tile of inputs staged once into LDS: 66 rows (64 + 2 halo) x 16 floats, per array.
    __shared__ __align__(16) float xs[(TILE_N + 2) * K_];
    __shared__ __align__(16) float ps[(TILE_N + 2) * K_];

    const int t  = threadIdx.x;
    const int b  = blockIdx.x / TILES_PER_B;
    const int n0 = (blockIdx.x % TILES_PER_B) * TILE_N;
    const int lr = 1 + (t >> 2);      // local row in LDS tile (1..64)
    const int kv = t & 3;             // which float4 of the 16-float row
    const int n  = n0 + (t >> 2);     // global row handled by this thread

    const uint64_t xbase = (uint64_t)(uintptr_t)x;
    const uint64_t pbase = (uint64_t)(uintptr_t)p;
    // Low 32 bits of a generic LDS address are the LDS byte offset (aperture rules).
    const uint32_t xs_base = (uint32_t)(uintptr_t)xs;
    const uint32_t ps_base = (uint32_t)(uintptr_t)ps;

    // Main tile: each thread DMAs its own (row, float4) element of x and param.
    {
        uint32_t voff = (uint32_t)((((size_t)b * N_ + n) * K_ + kv * 4) * sizeof(float));
        uint32_t loff = (uint32_t)((lr * K_ + kv * 4) * sizeof(float));
        async_load_b128(xs_base + loff, voff, xbase);
        async_load_b128(ps_base + loff, voff, pbase);
    }
    // Halo rows: local row 0 (global n0-1) and local row 65 (global n0+64),
    // clamped at the batch-slice edges (edge outputs bypass the stencil, so
    // clamped halo values are never consumed by a valid interior formula).
    if (t < 16) {
        const int top  = (t >> 2) & 1;            // 0 = bottom halo, 1 = top halo
        const int isp  = t >> 3;                  // 0 = x array, 1 = param array
        const int kvh  = t & 3;
        const int hl   = top ? (TILE_N + 1) : 0;  // local row in LDS
        int gn = n0 - 1 + hl;
        gn = gn < 0 ? 0 : (gn > N_ - 1 ? N_ - 1 : gn);
        uint32_t voff = (uint32_t)((((size_t)b * N_ + gn) * K_ + kvh * 4) * sizeof(float));
        uint32_t loff = (uint32_t)((hl * K_ + kvh * 4) * sizeof(float));
        if (isp == 0) async_load_b128(xs_base + loff, voff, xbase);
        else          async_load_b128(ps_base + loff, voff, pbase);
    }

    // Wait for our own async DMAs, then barrier so the whole tile is visible.
    asm volatile("s_wait_asynccnt 0" ::: "memory");
    __syncthreads();

    const f4 xm = *(const f4*)&xs[(lr - 1) * K_ + kv * 4];
    const f4 xc = *(const f4*)&xs[(lr    ) * K_ + kv * 4];
    const f4 xp = *(const f4*)&xs[(lr + 1) * K_ + kv * 4];
    const f4 pm = *(const f4*)&ps[(lr - 1) * K_ + kv * 4];
    const f4 pc = *(const f4*)&ps[(lr    ) * K_ + kv * 4];
    const f4 pp = *(const f4*)&ps[(lr + 1) * K_ + kv * 4];

    f4 r;
    r.x = convex_elem(xm.x, pm.x, xc.x, pc.x, xp.x, pp.x);
    r.y = convex_elem(xm.y, pm.y, xc.y, pc.y, xp.y, pp.y);
    r.z = convex_elem(xm.z, pm.z, xc.z, pc.z, xp.z, pp.z);
    r.w = convex_elem(xm.w, pm.w, xc.w, pc.w, xp.w, pp.w);

    // Zero-padded Dy at the slice edges => dy = 0 => out = y - param = x.
    if (n == 0 || n == N_ - 1) r = xc;

    // Streaming output: non-temporal store (th:TH_STORE_NT) so the 134 MB of
    // output doesn't evict the input lines that neighbor tiles re-read (halos).
    f4* dst = (f4*)&out[(((size_t)b * N_ + n) * K_ + kv * 4)];
    __builtin_nontemporal_store(r, dst);
}

extern "C" void kernel_launch(void* const* d_in, const int* in_sizes, int n_in,
                              void* d_out, int out_size, void* d_ws, size_t ws_size,
                              hipStream_t stream) {
    (void)in_sizes; (void)n_in; (void)out_size; (void)d_ws; (void)ws_size;
    const float* x = (const float*)d_in[0];
    const float* p = (const float*)d_in[1];
    float* out     = (float*)d_out;

    const int blocks = B_ * TILES_PER_B;   // 32768 workgroups of 256 threads (8 waves each)
    convex_stencil_kernel<<<blocks, 256, 0, stream>>>(x, p, out);
}